// ContextAwareModel_61306363183801
// MI455X (gfx1250) — compile-verified
//
#include <hip/hip_runtime.h>
#include <math.h>

#define EMB   300
#define HID   512
#define NSTEP 96
#define NGATE 4096   // 2 dirs * 4 gates * 512
#define NUNIT 1024   // 2 dirs * 512
#define NWG   64

typedef __attribute__((ext_vector_type(2))) float v2f;
typedef __attribute__((ext_vector_type(8))) float v8f;

// ---- workspace layout (float offsets) ----
#define X_OFF     0u                                   // 96*300      = 28800
#define WIHT_OFF  28800u                               // 300*4096    = 1228800
#define BIAS_OFF  (WIHT_OFF + 1228800u)                // 4096
#define ZX_OFF    (BIAS_OFF + 4096u)                   // 96*4096     = 393216
#define H_OFF     (ZX_OFF + 393216u)                   // 2*1024      = 2048
#define CNT_OFF   (H_OFF + 2048u)                      // 16
#define ENC_OFF   (CNT_OFF + 16u)                      // 96*1024     = 98304
// total ~1.76M floats ~= 7.0 MB

// ------------------- embeddings for batch row 0 -------------------
__global__ void k_embed(const int* __restrict__ tokens, const float* __restrict__ wm,
                        float* __restrict__ ws) {
    int idx = blockIdx.x * blockDim.x + threadIdx.x;
    if (idx >= NSTEP * EMB) return;
    int t = idx / EMB, e = idx - t * EMB;
    // input_tensor is (B,S) row-major; batch row 0 = first 96 ints
    ws[X_OFF + idx] = wm[(size_t)tokens[t] * EMB + e];
}

// ------------------- pack W_ih transposed: (2048 x 300) -> (300 x 4096) -------------------
__global__ void k_transpose(const float* __restrict__ src, float* __restrict__ dst,
                            int K, int colBase) {
    int idx = blockIdx.x * blockDim.x + threadIdx.x;  // K*2048 threads, exact grid
    int k = idx >> 11;
    int r = idx & 2047;
    dst[(size_t)k * NGATE + colBase + r] = src[(size_t)r * K + k];
}

__global__ void k_bias(const float* __restrict__ bi_f, const float* __restrict__ bh_f,
                       const float* __restrict__ bi_b, const float* __restrict__ bh_b,
                       float* __restrict__ dst) {
    int n = blockIdx.x * blockDim.x + threadIdx.x;    // 4096
    int r = n & 2047;
    const float* bi = (n < 2048) ? bi_f : bi_b;
    const float* bh = (n < 2048) ? bh_f : bh_b;
    dst[n] = bi[r] + bh[r];
}

// ------------------- input projection GEMM via fp32 WMMA -------------------
// Zx(96 x 4096) = X(96 x 300) @ WihT(300 x 4096) + bias
// One wave per 16x16 tile; 75 chunks of K=4; exact grid, branch-free (EXEC all ones).
__global__ void __launch_bounds__(256) k_inproj(float* __restrict__ ws) {
    const float* __restrict__ X    = ws + X_OFF;
    const float* __restrict__ Wt   = ws + WIHT_OFF;
    const float* __restrict__ bias = ws + BIAS_OFF;
    float* __restrict__ Zx         = ws + ZX_OFF;

    int wave = (blockIdx.x * blockDim.x + threadIdx.x) >> 5;   // 0..1535
    int lane = threadIdx.x & 31;
    int m0 = (wave % 6) * 16;
    int n0 = (wave / 6) * 16;
    // 32-bit A (16x4) layout: lanes 0-15 hold K=k0+{0,1}, lanes 16-31 hold K=k0+{2,3}
    int l  = lane & 15;
    int kk = (lane >> 4) << 1;

    v8f acc = {0.f, 0.f, 0.f, 0.f, 0.f, 0.f, 0.f, 0.f};
    const float* xrow = X  + (size_t)(m0 + l) * EMB + kk;        // A: row m0+l
    const float* wcol = Wt + (size_t)kk * NGATE + (n0 + l);      // B: col n0+l

    for (int kc = 0; kc < 75; ++kc) {
        v2f a, b;
        a[0] = xrow[kc * 4 + 0];
        a[1] = xrow[kc * 4 + 1];
        b[0] = wcol[(size_t)(kc * 4 + 0) * NGATE];
        b[1] = wcol[(size_t)(kc * 4 + 1) * NGATE];
        acc = __builtin_amdgcn_wmma_f32_16x16x4_f32(
            false, a, false, b, (short)0, acc, false, false);
    }
    // 32-bit C/D 16x16: VGPR r -> M=r (lanes 0-15) / M=8+r (lanes 16-31), N=lane&15
    int row = m0 + (lane >> 4) * 8;
    int col = n0 + l;
    float bv = bias[col];
#pragma unroll
    for (int r = 0; r < 8; ++r)
        Zx[(size_t)(row + r) * NGATE + col] = acc[r] + bv;
}

// ------------------- sequential LSTM recurrence, 64 persistent WGs, LDS-resident W_hh ----
// WG w: direction d = w>>5, units [ub, ub+16) -> 64 gate-rows, LDS slice 64x512 (padded 65).
// thread = (k-quarter = tid>>6, row_local = tid&63). Cell state c in registers of tid<16.
// h ping-pongs through global; one device-wide phase barrier per step.
__global__ void __launch_bounds__(256, 1) k_recur(float* __restrict__ ws,
                                                  const float* __restrict__ Whh_f,
                                                  const float* __restrict__ Whh_b) {
    extern __shared__ float smem[];
    float* Wlds  = smem;                       // 512 rows * 65 (64 + pad)
    float* h_lds = smem + 512 * 65;            // 512
    float* zp    = h_lds + 512;                // 4 x 64 partials
    float* zl    = zp + 256;                   // 64 combined pre-activations

    const float* __restrict__ Zx = ws + ZX_OFF;
    float* __restrict__ hbuf     = ws + H_OFF;     // 2 x 1024, zeroed at launch
    float* __restrict__ enc      = ws + ENC_OFF;   // 96 x 1024
    unsigned* cnt = (unsigned*)(ws + CNT_OFF);     // zeroed at launch

    int w   = blockIdx.x;       // 0..63
    int tid = threadIdx.x;      // 0..255
    int d   = w >> 5;           // direction (0 fwd, 1 "bwd")
    int ub  = (w & 31) * 16;    // unit base within direction
    int rl  = tid & 63;         // gate-row within WG: gate*16 + uu
    int kq  = tid >> 6;         // k-quarter 0..3
    int gate = rl >> 4, uu = rl & 15;
    int rcol = d * 2048 + gate * 512 + ub + uu;    // Zx column for this row
    const float* __restrict__ Whh = d ? Whh_b : Whh_f;

    // one-time preload of this WG's W_hh slice into LDS.
    // src reads: consecutive k -> coalesced; LDS writes: bank (k + r) % 64 -> conflict-free.
    for (int i = tid; i < 64 * 512; i += 256) {
        int r = i >> 9, k = i & 511;
        int g2 = r >> 4, u2 = r & 15;
        Wlds[k * 65 + r] = Whh[(size_t)(g2 * 512 + ub + u2) * 512 + k];
    }

    float c = 0.f;
    for (int t = 0; t < NSTEP; ++t) {
        int cur = t & 1;
        // broadcast previous h (this direction) into LDS
        h_lds[tid]       = hbuf[cur * NUNIT + d * HID + tid];
        h_lds[tid + 256] = hbuf[cur * NUNIT + d * HID + tid + 256];
        __syncthreads();   // also fences the Wlds preload on t==0

        float acc = 0.f;
        const float* wp = Wlds + (kq * 128) * 65 + rl;
        const float* hp = h_lds + kq * 128;
#pragma unroll 8
        for (int j = 0; j < 128; ++j)
            acc = fmaf(wp[j * 65], hp[j], acc);
        zp[kq * 64 + rl] = acc;
        __syncthreads();

        if (tid < 64) {    // combine k-quarters, add x-projection (incl. both biases)
            zl[tid] = zp[tid] + zp[64 + tid] + zp[128 + tid] + zp[192 + tid]
                    + Zx[(size_t)t * NGATE + rcol];
        }
        __syncthreads();

        if (tid < 16) {    // gating for this WG's 16 units
            float zi = zl[tid], zf = zl[16 + tid], zg = zl[32 + tid], zo = zl[48 + tid];
            float ig = 1.f / (1.f + expf(-zi));
            float fg = 1.f / (1.f + expf(-zf));
            float og = 1.f / (1.f + expf(-zo));
            float gg = tanhf(zg);
            c = fg * c + ig * gg;
            float h = og * tanhf(c);
            int u = d * HID + ub + tid;
            hbuf[(cur ^ 1) * NUNIT + u] = h;
            enc[(size_t)t * NUNIT + u]  = h;     // enc row = [hf | hb]
        }

        // device-wide phase barrier: target = NWG*(t+1)
        __threadfence();
        __syncthreads();
        if (tid == 0) {
            __hip_atomic_fetch_add(cnt, 1u, __ATOMIC_ACQ_REL, __HIP_MEMORY_SCOPE_AGENT);
            unsigned target = (unsigned)NWG * (unsigned)(t + 1);
            while (__hip_atomic_load(cnt, __ATOMIC_ACQUIRE, __HIP_MEMORY_SCOPE_AGENT) < target)
                __builtin_amdgcn_s_sleep(1);
        }
        __syncthreads();
    }
}

// ------------------- gather + linear + sigmoid -------------------
__global__ void k_out(const int* __restrict__ tgt, const float* __restrict__ Wout,
                      const float* __restrict__ bout, const float* __restrict__ enc,
                      float* __restrict__ out) {
    __shared__ float red[256];
    int b = blockIdx.x;
    const float* row = enc + (size_t)tgt[b] * NUNIT;
    float s = 0.f;
    for (int j = threadIdx.x; j < NUNIT; j += 256) s += row[j] * Wout[j];
    red[threadIdx.x] = s;
    __syncthreads();
    for (int off = 128; off > 0; off >>= 1) {
        if (threadIdx.x < off) red[threadIdx.x] += red[threadIdx.x + off];
        __syncthreads();
    }
    if (threadIdx.x == 0)
        out[b] = 1.f / (1.f + expf(-(red[0] + bout[0])));
}

extern "C" void kernel_launch(void* const* d_in, const int* in_sizes, int n_in,
                              void* d_out, int out_size, void* d_ws, size_t ws_size,
                              hipStream_t stream) {
    const int*   tokens = (const int*)d_in[0];
    const int*   tgt    = (const int*)d_in[1];
    const float* wm     = (const float*)d_in[3];
    const float* Wih_f  = (const float*)d_in[4];
    const float* Whh_f  = (const float*)d_in[5];
    const float* bih_f  = (const float*)d_in[6];
    const float* bhh_f  = (const float*)d_in[7];
    const float* Wih_b  = (const float*)d_in[8];
    const float* Whh_b  = (const float*)d_in[9];
    const float* bih_b  = (const float*)d_in[10];
    const float* bhh_b  = (const float*)d_in[11];
    const float* Wout   = (const float*)d_in[12];
    const float* bout   = (const float*)d_in[13];
    float* ws  = (float*)d_ws;
    float* out = (float*)d_out;

    // 64 rows * 512 k padded to stride 65, + h broadcast + partials + z
    const int recur_lds = (512 * 65 + 512 + 256 + 64) * (int)sizeof(float);  // 136448 B
    hipFuncSetAttribute((const void*)k_recur,
                        hipFuncAttributeMaxDynamicSharedMemorySize, recur_lds);

    // zero h ping-pong buffers + barrier counter (captured into the graph -> replay-safe)
    hipMemsetAsync(ws + H_OFF, 0, (2048u + 16u) * sizeof(float), stream);

    k_embed    <<<113, 256, 0, stream>>>(tokens, wm, ws);
    k_transpose<<<(300 * 2048) / 256, 256, 0, stream>>>(Wih_f, ws + WIHT_OFF, EMB, 0);
    k_transpose<<<(300 * 2048) / 256, 256, 0, stream>>>(Wih_b, ws + WIHT_OFF, EMB, 2048);
    k_bias     <<<16, 256, 0, stream>>>(bih_f, bhh_f, bih_b, bhh_b, ws + BIAS_OFF);
    k_inproj   <<<192, 256, 0, stream>>>(ws);          // 1536 waves = 6 x 256 WMMA tiles
    k_recur    <<<NWG, 256, recur_lds, stream>>>(ws, Whh_f, Whh_b);
    k_out      <<<96, 256, 0, stream>>>(tgt, Wout, bout, ws + ENC_OFF, out);
}